// Loss_yolov1_1194000908972
// MI455X (gfx1250) — compile-verified
//
#include <hip/hip_runtime.h>
#include <hip/hip_bf16.h>

typedef __attribute__((ext_vector_type(2))) float v2f;
typedef __attribute__((ext_vector_type(8))) float v8f;

#if defined(__has_builtin)
#if __has_builtin(__builtin_amdgcn_wmma_f32_16x16x4_f32)
#define HAVE_WMMA_F32_16X16X4 1
#endif
#endif

// Exact f32 wave32 reduction. With A = ones(16x4), D[m][n] = sum_k B[k][n]
// (independent of m and of the exact A/B lane layouts). B holds one partial
// per lane in VGPR0 and 0 in VGPR1, so the 64 B elements are exactly the 32
// lane partials plus 32 zeros. Each lane's d[0] = colsum(lane%16); summing the
// 16-lane group (xor 1,2,4,8) yields the full wave total in every lane.
__device__ __forceinline__ float wave_reduce(float p) {
#ifdef HAVE_WMMA_F32_16X16X4
    v2f a; a[0] = 1.0f; a[1] = 1.0f;
    v2f b; b[0] = p;    b[1] = 0.0f;
    v8f c = {};
    c = __builtin_amdgcn_wmma_f32_16x16x4_f32(
        /*neg_a=*/false, a, /*neg_b=*/false, b,
        /*c_mod=*/(short)0, c, /*reuse_a=*/false, /*reuse_b=*/false);
    float t = c[0];
#else
    float t = p;
    t += __shfl_xor(t, 16, 32);
#endif
    t += __shfl_xor(t, 1, 32);
    t += __shfl_xor(t, 2, 32);
    t += __shfl_xor(t, 4, 32);
    t += __shfl_xor(t, 8, 32);
    return t; // total in every lane
}

__device__ __forceinline__ float sq(float x) { return x * x; }

// IOU of predicted box (x,y,w,h in cell (i,j)) vs ground-truth box, both
// converted to xyxy exactly as the reference does (true divides by S=7).
__device__ __forceinline__ float iou_fn(float x, float y, float w, float h,
                                        float gx, float gy, float gw, float gh,
                                        float mi, float nj) {
    float cx  = (x + mi) / 7.0f,  cy  = (y + nj) / 7.0f;
    float bx1 = cx - 0.5f * w,    by1 = cy - 0.5f * h;
    float bx2 = cx + 0.5f * w,    by2 = cy + 0.5f * h;
    float gcx = (gx + mi) / 7.0f, gcy = (gy + nj) / 7.0f;
    float gx1 = gcx - 0.5f * gw,  gy1 = gcy - 0.5f * gh;
    float gx2 = gcx + 0.5f * gw,  gy2 = gcy + 0.5f * gh;
    float ix1 = fmaxf(bx1, gx1), iy1 = fmaxf(by1, gy1);
    float ix2 = fminf(bx2, gx2), iy2 = fminf(by2, gy2);
    float iw = fmaxf(ix2 - ix1, 0.0f), ih = fmaxf(iy2 - iy1, 0.0f);
    float inter = iw * ih;
    float a1 = (bx2 - bx1) * (by2 - by1);
    float a2 = (gx2 - gx1) * (gy2 - gy1);
    float denom = a1 + a2 - inter;
    return (inter > 0.0f) ? (inter / denom) : 0.0f;
}

#define NTL(p) __builtin_nontemporal_load(p)

__global__ __launch_bounds__(256)
void yolo_loss_main(const float* __restrict__ pred,
                    const float* __restrict__ lab,
                    float* __restrict__ ws, int nCells) {
    const int cell = blockIdx.x * 256 + threadIdx.x;
    float lossc = 0.0f;
    if (cell < nCells) {
        const int b = cell / 49;
        const int s = cell - b * 49;
        const int i = s / 7;          // 'm' index in reference (first spatial axis)
        const int j = s - i * 7;      // 'n' index
        const float mi = (float)i, nj = (float)j;
        const float* __restrict__ P = pred + (size_t)b * 1470 + s; // channel c at P[c*49]
        const float* __restrict__ L = lab  + (size_t)b * 1470 + s;

        // Box/conf channels (coalesced: lanes cover consecutive cells)
        float p0 = NTL(P + 0 * 49), p1 = NTL(P + 1 * 49), p2 = NTL(P + 2 * 49),
              p3 = NTL(P + 3 * 49), p4 = NTL(P + 4 * 49);
        float p5 = NTL(P + 5 * 49), p6 = NTL(P + 6 * 49), p7 = NTL(P + 7 * 49),
              p8 = NTL(P + 8 * 49), p9 = NTL(P + 9 * 49);
        float l0 = NTL(L + 0 * 49), l1 = NTL(L + 1 * 49), l2 = NTL(L + 2 * 49),
              l3 = NTL(L + 3 * 49), l4 = NTL(L + 4 * 49);
        float l5 = NTL(L + 5 * 49), l6 = NTL(L + 6 * 49), l7 = NTL(L + 7 * 49),
              l8 = NTL(L + 8 * 49);

        float obj   = (l4 == 1.0f) ? 1.0f : 0.0f;
        float iou1  = iou_fn(p0, p1, p2, p3, l0, l1, l2, l3, mi, nj);
        float iou2  = iou_fn(p5, p6, p7, p8, l0, l1, l2, l3, mi, nj);
        float resp1 = (iou1 >= iou2) ? 1.0f : 0.0f;
        float nresp = 1.0f - resp1;

        float coor1 = 5.0f * (sq(p0 - l0) + sq(p1 - l1) +
                              sq(__builtin_sqrtf(p2) - __builtin_sqrtf(l2)) +
                              sq(__builtin_sqrtf(p3) - __builtin_sqrtf(l3)));
        float coor2 = 5.0f * (sq(p5 - l5) + sq(p6 - l6) +
                              sq(__builtin_sqrtf(p7) - __builtin_sqrtf(l7)) +
                              sq(__builtin_sqrtf(p8) - __builtin_sqrtf(l8)));
        float objc1 = sq(p4 - iou1);
        float objc2 = sq(p9 - iou2);

        float cls = 0.0f;
        #pragma unroll
        for (int c = 10; c < 30; ++c) {
            float d = NTL(P + c * 49) - NTL(L + c * 49);
            cls = __builtin_fmaf(d, d, cls);
        }

        float coor       = resp1 * coor1 + nresp * coor2;
        float obj_conf   = resp1 * objc1 + nresp * objc2;
        float noobj_resp = 0.5f * (resp1 * objc2 + nresp * objc1);
        float noobj_cell = 0.5f * (p4 * p4 + p9 * p9);

        lossc = obj * (coor + obj_conf + noobj_resp + cls) + (1.0f - obj) * noobj_cell;
    }
    // EXEC reconverges to all-ones here (full 256-thread blocks): safe for WMMA.
    float t = wave_reduce(lossc);

    __shared__ float smem[8];
    const int lane = threadIdx.x & 31;
    const int wid  = threadIdx.x >> 5;
    if (lane == 0) smem[wid] = t;
    __syncthreads();
    if (threadIdx.x == 0) {
        float acc = 0.0f;
        #pragma unroll
        for (int w = 0; w < 8; ++w) acc += smem[w];
        ws[blockIdx.x] = acc;
    }
}

__global__ __launch_bounds__(32)
void yolo_loss_final(const float* __restrict__ ws, int n,
                     float* __restrict__ out, float invB) {
    const int lane = threadIdx.x;
    float p = 0.0f;
    for (int k = lane; k < n; k += 32) p += ws[k];
    float t = wave_reduce(p);  // EXEC all-ones after loop reconvergence
    if (lane == 0) out[0] = t * invB;
}

extern "C" void kernel_launch(void* const* d_in, const int* in_sizes, int n_in,
                              void* d_out, int out_size, void* d_ws, size_t ws_size,
                              hipStream_t stream) {
    const float* pred = (const float*)d_in[0];
    const float* lab  = (const float*)d_in[1];
    float* out = (float*)d_out;
    float* ws  = (float*)d_ws;

    const long long total = in_sizes[0];            // B * 30 * 49
    const int B      = (int)(total / (30 * 49));    // 16384
    const int nCells = B * 49;                      // 802816
    const int threads = 256;
    const int blocks  = (nCells + threads - 1) / threads;  // 3136

    yolo_loss_main<<<blocks, threads, 0, stream>>>(pred, lab, ws, nCells);
    yolo_loss_final<<<1, 32, 0, stream>>>(ws, blocks, out, 1.0f / (float)B);
}